// GatingNetwork_84928683311853
// MI455X (gfx1250) — compile-verified
//
#include <hip/hip_runtime.h>
#include <math.h>

typedef __attribute__((ext_vector_type(2))) float v2f;
typedef __attribute__((ext_vector_type(8))) float v8f;
typedef __attribute__((ext_vector_type(4))) unsigned int u32x4;
typedef __attribute__((ext_vector_type(8))) int i32x8;
typedef __attribute__((ext_vector_type(4))) int i32x4;

#define EPSF 1e-12f
#define E_DIM 64
#define KCHUNK 64
#define HPAD 68              // 64 + 4 DWORD TDM pad -> bank-conflict-free rows
#define TOK_PER_BLOCK 128    // 8 waves * 16 tokens

// ---------------------------------------------------------------------------
// TDM: issue a 2-D tensor_load_to_lds (fp32 elements), LDS row-padded by
// 4 DWORDs every 64 DWORDs (pad_interval code 5 = 64 DW, pad_amount code 3 =
// 4 DW), so a tile row of 64 floats lands at LDS stride 68 floats.
// This toolchain's builtin takes 6 args (g0, g1, g2, g3, g4, cpol).
// ---------------------------------------------------------------------------
__device__ __forceinline__ void tdm_load_2d_f32(unsigned lds_byte_off,
                                                const void* gptr,
                                                unsigned tensor_d0, unsigned tensor_d1,
                                                unsigned long long stride0,
                                                unsigned tile_d0, unsigned tile_d1) {
    unsigned long long ga = (unsigned long long)(uintptr_t)gptr;
    u32x4 g0;
    g0[0] = 1u;                                             // count=1, load-descriptor
    g0[1] = lds_byte_off;                                   // lds_addr
    g0[2] = (unsigned)(ga & 0xffffffffu);                   // global_addr[31:0]
    g0[3] = (unsigned)((ga >> 32) & 0x1ffffffu)             // global_addr[56:32]
          | 0x80000000u;                                    // type=2 (image)
    i32x8 g1;
    g1[0] = (int)((2u << 16)                                // data_size = 4B
                | (1u << 20)                                // pad_enable
                | (5u << 22)                                // pad_interval: 64 DWORDs
                | (3u << 25));                              // pad_amount: 4 DWORDs
    g1[1] = (int)((tensor_d0 & 0xffffu) << 16);             // tensor_dim0[15:0]
    g1[2] = (int)((tensor_d0 >> 16) | ((tensor_d1 & 0xffffu) << 16));
    g1[3] = (int)((tensor_d1 >> 16) | ((tile_d0 & 0xffffu) << 16));
    g1[4] = (int)(tile_d1 & 0xffffu);                       // tile_dim1, tile_dim2=0
    g1[5] = (int)(unsigned)(stride0 & 0xffffffffu);         // tensor_dim0_stride lo
    g1[6] = (int)(unsigned)((stride0 >> 32) & 0xffffu);     // stride hi, dim1_stride=0
    g1[7] = 0;
    i32x4 z4 = {0, 0, 0, 0};                                // 2-D: groups 2/3 unused
    i32x8 z8 = {0, 0, 0, 0, 0, 0, 0, 0};
    __builtin_amdgcn_tensor_load_to_lds(g0, g1, z4, z4, z8, 0);
}

// ---------------------------------------------------------------------------
// Kernel 1: column-normalize sim_matrix -> wn, sigmoid(gates) -> sig
// ---------------------------------------------------------------------------
__global__ __launch_bounds__(256)
void gn_prep_kernel(const float* __restrict__ sim, const float* __restrict__ gates,
                    float* __restrict__ wn, float* __restrict__ sig, int C, int E) {
    __shared__ float red[256];
    const int e = blockIdx.x;
    const int t = threadIdx.x;

    float s = 0.0f;
    for (int c = t; c < C; c += 256) {
        float v = sim[(size_t)c * E + e];
        s += v * v;
    }
    red[t] = s;
    __syncthreads();
    for (int off = 128; off > 0; off >>= 1) {
        if (t < off) red[t] += red[t + off];
        __syncthreads();
    }
    const float inv = 1.0f / fmaxf(sqrtf(red[0]), EPSF);
    for (int c = t; c < C; c += 256) {
        wn[(size_t)c * E + e] = sim[(size_t)c * E + e] * inv;
    }
    if (t == 0) sig[e] = 1.0f / (1.0f + __expf(-gates[e]));
}

// ---------------------------------------------------------------------------
// Kernel 2: TDM-staged fp32 WMMA GEMM + fused routing.
//   logits = (h @ wn)/||h|| - sigmoid(gates); relu mask; top-2 fallback;
//   masked softmax.  Per wave: 16 tokens x 64 experts (4 v8f accumulators).
// ---------------------------------------------------------------------------
__global__ __launch_bounds__(256)
void gn_main_kernel(const float* __restrict__ h, const float* __restrict__ wn,
                    const float* __restrict__ sig, const int* __restrict__ minex_p,
                    float* __restrict__ routing, float* __restrict__ logits_out,
                    float* __restrict__ mask_out, int N, int C, int E) {
    __shared__ __align__(16) float s_h[TOK_PER_BLOCK * HPAD]; // 34 KB, later row buf
    __shared__ __align__(16) float s_w[KCHUNK * HPAD];        // 17 KB
    __shared__ float s_norm[TOK_PER_BLOCK];

    const int tid  = threadIdx.x;
    const int wave = tid >> 5;
    const int lane = tid & 31;
    const int half = lane >> 4;          // A/B fragment K-half (K+{0,1} / K+{2,3})
    const int l16  = lane & 15;
    const int tok0 = blockIdx.x * TOK_PER_BLOCK + wave * 16;

    const unsigned lds_h = (unsigned)(uintptr_t)s_h;   // LDS byte offsets
    const unsigned lds_w = (unsigned)(uintptr_t)s_w;

    v8f acc[4] = {};
    float nsq = 0.0f;

    const int arow = (wave * 16 + l16) * HPAD + half * 2;  // lane's h-tile row base

    const int NC = C / KCHUNK;
    for (int c = 0; c < NC; ++c) {
        const int k0 = c * KCHUNK;
        if (tid < 32) {
            // h tile: rows [blk*128, +128), cols [k0, k0+64), row stride C
            tdm_load_2d_f32(lds_h, h + (size_t)blockIdx.x * TOK_PER_BLOCK * C + k0,
                            (unsigned)C, (unsigned)N, (unsigned long long)C,
                            KCHUNK, TOK_PER_BLOCK);
            // wn tile: rows [k0, k0+64), cols [0, 64), row stride E
            tdm_load_2d_f32(lds_w, wn + (size_t)k0 * E_DIM,
                            E_DIM, (unsigned)C, E_DIM,
                            E_DIM, KCHUNK);
            __builtin_amdgcn_s_wait_tensorcnt(0);
        }
        __syncthreads();

        #pragma unroll 4
        for (int ks = 0; ks < KCHUNK; ks += 4) {
            v2f a = *(const v2f*)(&s_h[arow + ks]);
            nsq += a.x * a.x + a.y * a.y;          // fused ||h||^2
            const int brow = (ks + half * 2) * HPAD + l16;
            #pragma unroll
            for (int t = 0; t < 4; ++t) {
                v2f b;
                b.x = s_w[brow + t * 16];
                b.y = s_w[brow + HPAD + t * 16];
                acc[t] = __builtin_amdgcn_wmma_f32_16x16x4_f32(
                    false, a, false, b, (short)0, acc[t], false, false);
            }
        }
        __syncthreads();   // all reads done before next chunk's TDM overwrites
    }

    // Row norms: lanes L and L+16 jointly hold token l16's sum of squares.
    nsq += __shfl_xor(nsq, 16);
    const float nrm = fmaxf(sqrtf(nsq), EPSF);
    if (half == 0) s_norm[wave * 16 + l16] = nrm;
    __syncthreads();

    float invn[8];
    #pragma unroll
    for (int j = 0; j < 8; ++j)
        invn[j] = 1.0f / s_norm[wave * 16 + j + half * 8];

    // Epilogue: scale, subtract sigmoid(gates); emit logits + padded LDS rows.
    #pragma unroll
    for (int t = 0; t < 4; ++t) {
        const int col = t * 16 + l16;
        const float sg = sig[col];
        #pragma unroll
        for (int j = 0; j < 8; ++j) {
            const int m = j + half * 8;
            const float lg = acc[t][j] * invn[j] - sg;
            logits_out[(size_t)(tok0 + m) * E + col] = lg;
            s_h[(wave * 16 + m) * (E_DIM + 1) + col] = lg;   // reuse h-tile LDS
        }
    }
    __syncthreads();

    // Routing: lane < 16 handles one token's full 64-expert row.
    if (lane < 16) {
        const int m = wave * 16 + lane;
        const int token = blockIdx.x * TOK_PER_BLOCK + m;
        const float* row = &s_h[m * (E_DIM + 1)];
        const int k = *minex_p;                              // min_experts (== 2)

        float sumexp = 0.0f; int cnt = 0;
        float max1 = -INFINITY, max2 = -INFINITY; int i1 = 0, i2 = 1;
        for (int e = 0; e < E_DIM; ++e) {
            const float x = row[e];
            if (x > 0.0f) { cnt++; sumexp += __expf(x); }
            if (x > max1)      { max2 = max1; i2 = i1; max1 = x; i1 = e; }
            else if (x > max2) { max2 = x; i2 = e; }
        }
        const float inv_se = (cnt > 0) ? (1.0f / sumexp) : 0.0f;
        const float fb_w = 1.0f / (float)k;
        for (int e = 0; e < E_DIM; ++e) {
            const float x = row[e];
            float mk, rw;
            if (cnt > 0) {
                mk = (x > 0.0f) ? 1.0f : 0.0f;
                rw = (x > 0.0f) ? __expf(x) * inv_se : 0.0f;
            } else {
                const bool sel = (e == i1) || (e == i2);
                mk = sel ? 1.0f : 0.0f;
                rw = sel ? fb_w : 0.0f;
            }
            routing[(size_t)token * E + e] = rw;
            mask_out[(size_t)token * E + e] = mk;
        }
    }
}

// ---------------------------------------------------------------------------
extern "C" void kernel_launch(void* const* d_in, const int* in_sizes, int n_in,
                              void* d_out, int out_size, void* d_ws, size_t ws_size,
                              hipStream_t stream) {
    const float* h     = (const float*)d_in[0];   // [B*T, C] fp32
    const float* sim   = (const float*)d_in[1];   // [C, E]   fp32
    const float* gates = (const float*)d_in[2];   // [E]      fp32
    const int*   minex = (const int*)d_in[3];     // scalar

    const int E = in_sizes[2];                    // 64
    const int C = in_sizes[1] / E;                // 2048
    const int N = in_sizes[0] / C;                // 16384

    float* wn  = (float*)d_ws;                    // [C, E] normalized columns
    float* sig = wn + (size_t)C * E;              // [E] sigmoid(gates)

    float* routing = (float*)d_out;               // [N, E]
    float* logits  = routing + (size_t)N * E;     // [N, E]
    float* maskp   = logits + (size_t)N * E;      // [N, E]

    gn_prep_kernel<<<E, 256, 0, stream>>>(sim, gates, wn, sig, C, E);
    gn_main_kernel<<<N / TOK_PER_BLOCK, 256, 0, stream>>>(
        h, wn, sig, minex, routing, logits, maskp, N, C, E);
}